// WeightedProcrustes_86242943304132
// MI455X (gfx1250) — compile-verified
//
#include <hip/hip_runtime.h>
#include <hip/hip_bf16.h>

typedef __attribute__((ext_vector_type(2))) float v2f;
typedef __attribute__((ext_vector_type(8))) float v8f;

#define WPB   8     // waves per block (256 threads, wave32)
#define PARTS 40    // blocks per batch for stage 1

// ---------------------------------------------------------------------------
// Workspace layout: ws[b*16 + i*4 + j] accumulates the homogeneous outer
// product  C[i][j] = sum_m [a_i;1]*[b_j;1]  ->  Sab(3x3), Sa(col3), Sb(row3),
// count(3,3).
// ---------------------------------------------------------------------------

__global__ __launch_bounds__(256) void procrustes_zero_ws(float* ws, int n) {
  int i = blockIdx.x * blockDim.x + threadIdx.x;
  if (i < n) ws[i] = 0.0f;
}

// Stage 1: streaming gather + WMMA cross-covariance accumulation.
__global__ __launch_bounds__(256) void procrustes_stage1(
    const float* __restrict__ src, const float* __restrict__ tgt,
    float* __restrict__ ws, int N, int ns) {
  // wave-private transposed staging: [wave][src/ref][component(x,y,z,valid)][point]
  __shared__ float lds[WPB][2][4][32];

  const int batch = blockIdx.x;
  const int part  = blockIdx.y;
  const int lane  = threadIdx.x & 31;
  const int wave  = threadIdx.x >> 5;
  const int wavesPerBatch = gridDim.y * WPB;
  const int waveId  = part * WPB + wave;
  const int nChunks = (ns + 31) >> 5;
  const float scale = (float)(N - 1) / (float)(ns - 1);

  const float* sb = src + (size_t)batch * N * 3;
  const float* tb = tgt + (size_t)batch * N * 3;

  float (*msrc)[32] = lds[wave][0];
  float (*mref)[32] = lds[wave][1];

  v8f acc = {0.f, 0.f, 0.f, 0.f, 0.f, 0.f, 0.f, 0.f};

  // Fragment addressing (doc layout, K = 2*half + v, row/col = lane%16):
  // lanes with row/col >= 4 load duplicate data (m&3) -- rows/cols 4..15 of C
  // are never read, so no predication / zero-fill is needed and EXEC stays
  // all-ones throughout.
  const int half = lane >> 4;
  const int mm   = lane & 3;            // (lane & 15) clamped into the live rows
  const float* afrag = &msrc[mm][2 * half];
  const float* bfrag = &mref[mm][2 * half];

  for (int c = waveId; c < nChunks; c += wavesPerBatch) {
    const int s = (c << 5) + lane;      // sample index handled by this lane
    float sx = 0.f, sy = 0.f, sz = 0.f, rx = 0.f, ry = 0.f, rz = 0.f, valid = 0.f;
    if (s < ns) {
      int idx = (int)((float)s * scale);          // mimics linspace(...).astype(int32)
      if (idx > N - 1) idx = N - 1;
      const float* sp = sb + (size_t)idx * 3;
      const float* tp = tb + (size_t)idx * 3;
      sx = sp[0]; sy = sp[1]; sz = sp[2];
      rx = tp[0]; ry = tp[1]; rz = tp[2];
      valid = 1.0f;                      // homogeneous component: makes the WMMA
    }                                    // also accumulate Sa, Sb and the count
    // transpose into LDS (component-major: fragments are aligned float2 pairs)
    msrc[0][lane] = sx; msrc[1][lane] = sy; msrc[2][lane] = sz; msrc[3][lane] = valid;
    mref[0][lane] = rx; mref[1][lane] = ry; mref[2][lane] = rz; mref[3][lane] = valid;
    __builtin_amdgcn_wave_barrier();     // wave-private region: ordering only

    // 8 WMMAs, 4 points each; C accumulates across the whole chunk loop.
    #pragma unroll
    for (int k0 = 0; k0 < 32; k0 += 4) {
      v2f av = *(const v2f*)(afrag + k0);   // ds_load_b64
      v2f bv = *(const v2f*)(bfrag + k0);   // ds_load_b64
      acc = __builtin_amdgcn_wmma_f32_16x16x4_f32(
          /*neg_a=*/false, av, /*neg_b=*/false, bv,
          /*c_mod=*/(short)0, acc, /*reuse_a=*/false, /*reuse_b=*/false);
    }
    __builtin_amdgcn_wave_barrier();
  }

  // C layout: VGPR r, lanes 0-15 -> M=r, N=lane.  Flush the 4x4 corner.
  if (lane < 4) {
    float* w = ws + batch * 16;
    atomicAdd(&w[0 * 4 + lane], acc[0]);
    atomicAdd(&w[1 * 4 + lane], acc[1]);
    atomicAdd(&w[2 * 4 + lane], acc[2]);
    atomicAdd(&w[3 * 4 + lane], acc[3]);
  }
}

// ---------------------------------------------------------------------------
// Stage 2: per-batch 3x3 SVD (Jacobi on H^T H) + Kabsch assembly.
// ---------------------------------------------------------------------------

__device__ __forceinline__ float det3(const float M[3][3]) {
  return M[0][0] * (M[1][1] * M[2][2] - M[1][2] * M[2][1])
       - M[0][1] * (M[1][0] * M[2][2] - M[1][2] * M[2][0])
       + M[0][2] * (M[1][0] * M[2][1] - M[1][1] * M[2][0]);
}

__device__ __forceinline__ void jrot(float A[3][3], float V[3][3], int p, int q) {
  float apq = A[p][q];
  if (fabsf(apq) < 1e-20f) return;
  float app = A[p][p], aqq = A[q][q];
  float tau = (aqq - app) / (2.0f * apq);
  float t = copysignf(1.0f, tau) / (fabsf(tau) + sqrtf(1.0f + tau * tau));
  float cc = 1.0f / sqrtf(1.0f + t * t);
  float ss = t * cc;
  int r = 3 - p - q;
  float arp = A[r][p], arq = A[r][q];
  A[r][p] = A[p][r] = cc * arp - ss * arq;
  A[r][q] = A[q][r] = ss * arp + cc * arq;
  A[p][p] = app - t * apq;
  A[q][q] = aqq + t * apq;
  A[p][q] = A[q][p] = 0.0f;
  #pragma unroll
  for (int i = 0; i < 3; i++) {
    float vip = V[i][p], viq = V[i][q];
    V[i][p] = cc * vip - ss * viq;
    V[i][q] = ss * vip + cc * viq;
  }
}

__global__ void procrustes_stage2(const float* __restrict__ ws,
                                  float* __restrict__ out, int B_) {
  int b = blockIdx.x * blockDim.x + threadIdx.x;
  if (b >= B_) return;
  const float* w = ws + b * 16;

  float cnt = w[15];
  float inv = (cnt > 0.f) ? (1.0f / cnt) : 0.0f;
  float ac[3], bc[3], H[3][3];
  #pragma unroll
  for (int i = 0; i < 3; i++) { ac[i] = w[i * 4 + 3] * inv; bc[i] = w[12 + i] * inv; }
  #pragma unroll
  for (int i = 0; i < 3; i++)
    #pragma unroll
    for (int j = 0; j < 3; j++) {
      float h = w[i * 4 + j] - w[i * 4 + 3] * w[12 + j] * inv;  // Sab - Sa Sb^T / M
      H[i][j] = (h == h) ? h : 0.0f;                            // NaN -> 0
    }

  // A = H^T H, Jacobi eigen-decomposition -> V, eigenvalues
  float A[3][3];
  #pragma unroll
  for (int i = 0; i < 3; i++)
    #pragma unroll
    for (int j = 0; j < 3; j++) {
      float s = 0.f;
      #pragma unroll
      for (int k = 0; k < 3; k++) s += H[k][i] * H[k][j];
      A[i][j] = s;
    }
  float V[3][3] = {{1, 0, 0}, {0, 1, 0}, {0, 0, 1}};
  for (int sweep = 0; sweep < 12; sweep++) {
    jrot(A, V, 0, 1); jrot(A, V, 0, 2); jrot(A, V, 1, 2);
  }
  float lam[3] = {A[0][0], A[1][1], A[2][2]};
  // sort columns descending (matches jnp.linalg.svd ordering)
  for (int a2 = 0; a2 < 2; a2++)
    for (int b2 = a2 + 1; b2 < 3; b2++)
      if (lam[b2] > lam[a2]) {
        float tl = lam[a2]; lam[a2] = lam[b2]; lam[b2] = tl;
        for (int i = 0; i < 3; i++) { float tv = V[i][a2]; V[i][a2] = V[i][b2]; V[i][b2] = tv; }
      }

  // U columns: u_k = normalize(GS(H v_k)) — sign preserved so det(U) is the SVD's
  float U[3][3];
  for (int k = 0; k < 3; k++) {
    float wv[3];
    for (int i = 0; i < 3; i++) {
      float s = 0.f;
      for (int j = 0; j < 3; j++) s += H[i][j] * V[j][k];
      wv[i] = s;
    }
    for (int kk = 0; kk < k; kk++) {
      float d = wv[0] * U[0][kk] + wv[1] * U[1][kk] + wv[2] * U[2][kk];
      for (int i = 0; i < 3; i++) wv[i] -= d * U[i][kk];
    }
    float n2 = wv[0] * wv[0] + wv[1] * wv[1] + wv[2] * wv[2];
    if (n2 <= 1e-24f) {
      // degenerate fallback: any axis orthogonal to previous columns
      for (int t = 0; t < 3 && n2 <= 1e-24f; t++) {
        wv[0] = (t == 0) ? 1.f : 0.f; wv[1] = (t == 1) ? 1.f : 0.f; wv[2] = (t == 2) ? 1.f : 0.f;
        for (int kk = 0; kk < k; kk++) {
          float d = wv[0] * U[0][kk] + wv[1] * U[1][kk] + wv[2] * U[2][kk];
          for (int i = 0; i < 3; i++) wv[i] -= d * U[i][kk];
        }
        n2 = wv[0] * wv[0] + wv[1] * wv[1] + wv[2] * wv[2];
      }
    }
    float rn = 1.0f / sqrtf(n2);
    for (int i = 0; i < 3; i++) U[i][k] = wv[i] * rn;
  }

  float prod = det3(U) * det3(V);
  float sgn = (prod > 0.f) ? 1.f : ((prod < 0.f) ? -1.f : 0.f);
  float d3[3] = {1.f, 1.f, sgn};

  float R[3][3];
  #pragma unroll
  for (int i = 0; i < 3; i++)
    #pragma unroll
    for (int j = 0; j < 3; j++) {
      float s = 0.f;
      #pragma unroll
      for (int k = 0; k < 3; k++) s += d3[k] * V[i][k] * U[j][k];
      R[i][j] = s;
    }
  float t[3];
  #pragma unroll
  for (int i = 0; i < 3; i++) {
    float s = 0.f;
    #pragma unroll
    for (int j = 0; j < 3; j++) s += R[i][j] * ac[j];
    t[i] = bc[i] - s;
  }

  float* o = out + b * 16;
  #pragma unroll
  for (int r = 0; r < 3; r++) {
    o[r * 4 + 0] = R[r][0]; o[r * 4 + 1] = R[r][1]; o[r * 4 + 2] = R[r][2];
    o[r * 4 + 3] = t[r];
  }
  o[12] = 0.f; o[13] = 0.f; o[14] = 0.f; o[15] = 1.f;
}

// ---------------------------------------------------------------------------

extern "C" void kernel_launch(void* const* d_in, const int* in_sizes, int n_in,
                              void* d_out, int out_size, void* d_ws, size_t ws_size,
                              hipStream_t stream) {
  const float* src = (const float*)d_in[0];   // (B, N, 3) f32
  const float* tgt = (const float*)d_in[1];   // (B, N, 3) f32
  float* out = (float*)d_out;                 // (B, 4, 4) f32
  float* ws  = (float*)d_ws;                  // B*16 f32 accumulators

  const int B_ = 32;
  const int N  = in_sizes[0] / (B_ * 3);
  const int ns = N / 5;

  procrustes_zero_ws<<<dim3((B_ * 16 + 255) / 256), dim3(256), 0, stream>>>(ws, B_ * 16);
  procrustes_stage1<<<dim3(B_, PARTS), dim3(256), 0, stream>>>(src, tgt, ws, N, ns);
  procrustes_stage2<<<dim3(1), dim3(32), 0, stream>>>(ws, out, B_);
}